// CalculatorBaseTorch_3607772529447
// MI455X (gfx1250) — compile-verified
//
#include <hip/hip_runtime.h>
#include <hip/hip_bf16.h>

// ---------------------------------------------------------------------------
// Symmetric edge scatter-add (erfc-screened Coulomb), MI455X / gfx1250.
// Memory/atomic-bound: 12 B/edge streamed, 2x16 B L2-hot gathers, 8 no-return
// global_atomic_add_f32 per edge. CDNA5 path: edge streams staged into LDS
// with global_load_async_to_lds_b128 (4 edges/thread per chunk), double
// buffered, synchronized with s_wait_asynccnt.
// ---------------------------------------------------------------------------

#define AS1 __attribute__((address_space(1)))
#define AS3 __attribute__((address_space(3)))

typedef int v4i __attribute__((ext_vector_type(4)));

#define CHUNK 256            // threads per block
#define EPT 4                // edges per thread per chunk
#define CEDGES (CHUNK * EPT) // 1024 edges per block-chunk
#define MAX_BLOCKS 1024      // persistent blocks grid-striding over chunks

#if __has_builtin(__builtin_amdgcn_global_load_async_to_lds_b128)
#define USE_ASYNC_LDS 1
#else
#define USE_ASYNC_LDS 0
#endif

#if USE_ASYNC_LDS
__device__ __forceinline__ void wait_async_le3() {
#if __has_builtin(__builtin_amdgcn_s_wait_asynccnt)
  __builtin_amdgcn_s_wait_asynccnt(3);
#else
  asm volatile("s_wait_asynccnt 0x3" ::: "memory");
#endif
  asm volatile("" ::: "memory");  // keep LDS reads below the wait
}
__device__ __forceinline__ void wait_async_le0() {
#if __has_builtin(__builtin_amdgcn_s_wait_asynccnt)
  __builtin_amdgcn_s_wait_asynccnt(0);
#else
  asm volatile("s_wait_asynccnt 0x0" ::: "memory");
#endif
  asm volatile("" ::: "memory");
}
#endif

__device__ __forceinline__ void process_edge(int ai, int aj, float d,
                                             const float4* __restrict__ charges,
                                             float* __restrict__ out) {
  // 0.5 factor (reference's final /2) folded into pot.
  const float pot = 0.5f * erfcf(d * 0.70710678118654752440f) / d;
  const float4 qi = charges[ai];
  const float4 qj = charges[aj];
  float* oi = out + 4ll * ai;
  float* oj = out + 4ll * aj;
  // Results discarded -> no-return GLOBAL_ATOMIC_ADD_F32 (L2 atomic units).
  atomicAdd(oi + 0, qj.x * pot);
  atomicAdd(oi + 1, qj.y * pot);
  atomicAdd(oi + 2, qj.z * pot);
  atomicAdd(oi + 3, qj.w * pot);
  atomicAdd(oj + 0, qi.x * pot);
  atomicAdd(oj + 1, qi.y * pot);
  atomicAdd(oj + 2, qi.z * pot);
  atomicAdd(oj + 3, qi.w * pot);
}

__global__ void __launch_bounds__(CHUNK) zero_f32_kernel(float* __restrict__ p, int n) {
  int i = blockIdx.x * blockDim.x + threadIdx.x;
  if (i < n) p[i] = 0.0f;
}

__global__ void __launch_bounds__(CHUNK) edge_scatter_kernel(
    const int* __restrict__ nbr,        // [nEdges*2] (i,j) pairs
    const float* __restrict__ dist,     // [nEdges]
    const float4* __restrict__ charges, // [nAtoms]
    float* __restrict__ out,            // [nAtoms*4], pre-zeroed
    int nEdges) {
  const int tid = threadIdx.x;
  const int nChunks = (nEdges + CEDGES - 1) / CEDGES;

#if USE_ASYNC_LDS
  // Double-buffered LDS staging: each thread owns a 32B idx slice + 16B dist
  // slice per buffer, filled by its own async b128 loads. Each wave consumes
  // only slots written by its own lanes, so ASYNCcnt (per-wave, in-order load
  // completion) is the only synchronization needed — no workgroup barrier.
  __shared__ __align__(16) int   sIdx[2][CHUNK * 2 * EPT];  // 8 ints/thread
  __shared__ __align__(16) float sDist[2][CHUNK * EPT];     // 4 floats/thread

  int c = blockIdx.x;
  if (c >= nChunks) return;

  auto chunk_full = [&](int ck) -> bool {
    return (long long)ck * CEDGES + CEDGES <= (long long)nEdges;
  };
  auto issue = [&](int ck, int b) {
    const long long eb = (long long)ck * CEDGES + EPT * tid;  // first edge
    __builtin_amdgcn_global_load_async_to_lds_b128(
        (AS1 v4i*)(nbr + 2 * eb),     (AS3 v4i*)&sIdx[b][8 * tid],     0, 0);
    __builtin_amdgcn_global_load_async_to_lds_b128(
        (AS1 v4i*)(nbr + 2 * eb + 4), (AS3 v4i*)&sIdx[b][8 * tid + 4], 0, 0);
    __builtin_amdgcn_global_load_async_to_lds_b128(
        (AS1 v4i*)(dist + eb),        (AS3 v4i*)&sDist[b][4 * tid],    0, 0);
  };
  auto consume_lds = [&](int b) {
    const int4   p01 = *(const int4*)&sIdx[b][8 * tid];
    const int4   p23 = *(const int4*)&sIdx[b][8 * tid + 4];
    const float4 dv  = *(const float4*)&sDist[b][4 * tid];
    process_edge(p01.x, p01.y, dv.x, charges, out);
    process_edge(p01.z, p01.w, dv.y, charges, out);
    process_edge(p23.x, p23.y, dv.z, charges, out);
    process_edge(p23.z, p23.w, dv.w, charges, out);
  };
  auto process_direct = [&](int ck) {  // partial final chunk: guarded loads
    const long long eb = (long long)ck * CEDGES + EPT * tid;
#pragma unroll
    for (int k = 0; k < EPT; ++k) {
      const long long e = eb + k;
      if (e < nEdges) {
        const int2 p = ((const int2*)nbr)[e];
        process_edge(p.x, p.y, dist[e], charges, out);
      }
    }
  };

  int buf = 0;
  bool curA = chunk_full(c);
  if (curA) issue(c, 0);  // prologue

  for (; c < nChunks; c += gridDim.x) {
    const int  cn    = c + gridDim.x;
    const bool nextA = (cn < nChunks) && chunk_full(cn);
    if (nextA) issue(cn, buf ^ 1);
    if (curA) {
      if (nextA) wait_async_le3();  // current chunk's 3 async ops retired
      else       wait_async_le0();
      consume_lds(buf);
    } else {
      process_direct(c);
    }
    buf ^= 1;
    curA = nextA;
  }
#else
  // Fallback: plain coalesced streaming loads (still near bandwidth-optimal).
  long long e = (long long)blockIdx.x * blockDim.x + threadIdx.x;
  const long long stride = (long long)gridDim.x * blockDim.x;
  for (; e < nEdges; e += stride) {
    const int2  p = ((const int2*)nbr)[e];
    const float d = dist[e];
    process_edge(p.x, p.y, d, charges, out);
  }
#endif
}

extern "C" void kernel_launch(void* const* d_in, const int* in_sizes, int n_in,
                              void* d_out, int out_size, void* d_ws, size_t ws_size,
                              hipStream_t stream) {
  (void)n_in; (void)d_ws; (void)ws_size;
  const float* charges = (const float*)d_in[0];
  const int*   nbr     = (const int*)d_in[1];
  const float* dist    = (const float*)d_in[2];
  float*       out     = (float*)d_out;
  const int nEdges = in_sizes[2];

  // d_out is poisoned (0xAA) and not re-poisoned between replays: zero it
  // ourselves on the same stream before the atomic accumulation.
  {
    const int zb = (out_size + CHUNK - 1) / CHUNK;
    zero_f32_kernel<<<zb, CHUNK, 0, stream>>>(out, out_size);
  }

  const int nChunks = (nEdges + CEDGES - 1) / CEDGES;
  int grid = nChunks < MAX_BLOCKS ? nChunks : MAX_BLOCKS;
  if (grid < 1) grid = 1;
  edge_scatter_kernel<<<grid, CHUNK, 0, stream>>>(nbr, dist, (const float4*)charges,
                                                  out, nEdges);
}